// NextTokenModel_6502580486536
// MI455X (gfx1250) — compile-verified
//
#include <hip/hip_runtime.h>
#include <math.h>

// Problem constants (from reference)
#define H_   1024
#define NH_  16
#define DH_  64
#define L_   4
#define K_   128
#define N_   65536
#define S_   258     // 2*K+2
#define SK_  288     // attention row padded to multiple of KC_ (32) for the chunked K-loop

// GEMM tiling
#define KC_    32            // K chunk staged per TDM transfer
#define APAD_  36            // LDS row stride for K-chunk tiles (32 + 4 pad -> conflict-free)
#define A_LDS  (64 * APAD_)  // 2304 floats
#define B_LDS  (128 * APAD_) // 4608 floats (BT layout); NN layout needs 32*128=4096 <= this
#define BUF_LDS (A_LDS + B_LDS)

typedef float v2f __attribute__((ext_vector_type(2)));
typedef float v8f __attribute__((ext_vector_type(8)));
typedef unsigned int u32x4 __attribute__((ext_vector_type(4)));
typedef int i32x4 __attribute__((ext_vector_type(4)));
typedef int i32x8 __attribute__((ext_vector_type(8)));

__device__ __forceinline__ float gelu_f(float v) {
  return 0.5f * v * (1.0f + erff(v * 0.70710678118654752440f));
}

// ---------------------------------------------------------------------------
// TDM 2D tile load: global -> LDS (CDNA5 ISA 8.3/8.4 descriptor layout)
//  - tdim0/tdim1: VALID extent (elements/rows); OOB reads zero-fill the tile
//  - tile0/tile1: tile shape actually written to LDS
//  - pad: insert 4 dwords after every 32 dwords (row stride 36 floats in LDS)
// amdgpu-toolchain (clang-23) 6-arg builtin form.
// ---------------------------------------------------------------------------
__device__ __forceinline__ void tdm_load_2d(unsigned lds_addr, const void* gptr,
                                            unsigned tdim0, unsigned tdim1,
                                            unsigned tile0, unsigned tile1,
                                            unsigned row_stride_elems, bool pad) {
  unsigned long long ga = (unsigned long long)(size_t)gptr;
  u32x4 g0;
  g0[0] = 1u;                                  // count=1 (valid descriptor)
  g0[1] = lds_addr;                            // lds_addr [63:32]
  g0[2] = (unsigned)ga;                        // global_addr lo  [95:64]
  g0[3] = (unsigned)((ga >> 32) & 0x01FFFFFFu) // global_addr hi  [120:96]
        | 0x80000000u;                         // type=2 ("image") [127:126]
  i32x8 g1;
  unsigned dw0 = (2u << 16);                   // data_size = 4 bytes
  if (pad) dw0 |= (1u << 20)                   // pad_enable
              | (4u << 22)                     // pad_interval: 32 dwords
              | (3u << 25);                    // pad_amount: 4 dwords
  g1[0] = (int)dw0;
  g1[1] = (int)((tdim0 & 0xFFFFu) << 16);                    // tensor_dim0 lo16 [63:48]
  g1[2] = (int)((tdim0 >> 16) | ((tdim1 & 0xFFFFu) << 16));  // tdim0 hi / tdim1 lo
  g1[3] = (int)((tdim1 >> 16) | (tile0 << 16));              // tdim1 hi / tile_dim0
  g1[4] = (int)(tile1 & 0xFFFFu);                            // tile_dim1 (tile_dim2=0)
  g1[5] = (int)row_stride_elems;                             // tensor_dim0_stride lo32
  g1[6] = 0;
  g1[7] = 0;
  i32x4 z4 = {0, 0, 0, 0};
  i32x8 z8 = {0, 0, 0, 0, 0, 0, 0, 0};
  __builtin_amdgcn_tensor_load_to_lds(g0, g1, z4, z4, z8, 0);
}

// ---------------------------------------------------------------------------
// ||x|| (single block)
// ---------------------------------------------------------------------------
__global__ __launch_bounds__(256) void xnorm_kernel(const float* __restrict__ x,
                                                    float* __restrict__ xn) {
  __shared__ float red[256];
  int t = threadIdx.x;
  float s = 0.f;
  for (int i = t; i < H_; i += 256) { float v = x[i]; s += v * v; }
  red[t] = s; __syncthreads();
  for (int o = 128; o > 0; o >>= 1) { if (t < o) red[t] += red[t + o]; __syncthreads(); }
  if (t == 0) xn[0] = fmaxf(sqrtf(red[0]), 1e-8f);
}

// ---------------------------------------------------------------------------
// sim[i] = dot(mem[i], x) / (max(||mem[i]||,eps) * max(||x||,eps))
// one wave32 per row (bandwidth-bound: 256 MB streamed once)
// ---------------------------------------------------------------------------
__global__ __launch_bounds__(256) void sim_kernel(const float* __restrict__ mem,
                                                  const float* __restrict__ x,
                                                  const float* __restrict__ xnorm,
                                                  float* __restrict__ sim) {
  int wid = blockIdx.x * 8 + (threadIdx.x >> 5);
  int lane = threadIdx.x & 31;
  if (wid >= N_) return;
  const float* row = mem + (long)wid * H_;
  float dot = 0.f, nrm = 0.f;
  for (int i = lane; i < H_; i += 32) { float m = row[i]; dot += m * x[i]; nrm += m * m; }
#pragma unroll
  for (int off = 16; off >= 1; off >>= 1) {
    dot += __shfl_xor(dot, off, 32);
    nrm += __shfl_xor(nrm, off, 32);
  }
  if (lane == 0) sim[wid] = dot / (fmaxf(sqrtf(nrm), 1e-8f) * xnorm[0]);
}

// ---------------------------------------------------------------------------
__global__ void copy3_kernel(const float* __restrict__ a, const float* __restrict__ b,
                             const float* __restrict__ c, float* __restrict__ s) {
  int i = blockIdx.x * 256 + threadIdx.x;
  s[i] = a[i]; s[N_ + i] = b[i]; s[2 * N_ + i] = c[i];
}

// ---------------------------------------------------------------------------
// bitonic sort pass; blockIdx.y selects which of the 3 arrays
// ---------------------------------------------------------------------------
__global__ void bitonic_pass(float* __restrict__ base, int k, int j) {
  float* d = base + (long)blockIdx.y * N_;
  int i = blockIdx.x * 256 + threadIdx.x;
  int p = i ^ j;
  if (p > i) {
    bool asc = (i & k) == 0;
    float a = d[i], b = d[p];
    if ((a > b) == asc) { d[i] = b; d[p] = a; }
  }
}

// ---------------------------------------------------------------------------
// tie-aware rank score: ((lower+upper-1)/2)/N, squared, summed over 3 arrays
// ---------------------------------------------------------------------------
__device__ __forceinline__ float rank_comp(const float* __restrict__ srt, float v) {
  int lo = 0, hi = N_;
  while (lo < hi) { int m = (lo + hi) >> 1; if (srt[m] < v) lo = m + 1; else hi = m; }
  int left = lo;
  lo = 0; hi = N_;
  while (lo < hi) { int m = (lo + hi) >> 1; if (srt[m] <= v) lo = m + 1; else hi = m; }
  int right = lo;
  float r = ((float)(left + right - 1) * 0.5f) / (float)N_;
  return r * r;
}

__global__ void rank_kernel(const float* __restrict__ sur, const float* __restrict__ sim,
                            const float* __restrict__ noi, const float* __restrict__ sorted,
                            float* __restrict__ scores) {
  int i = blockIdx.x * 256 + threadIdx.x;
  float s = rank_comp(sorted, sur[i])
          + rank_comp(sorted + N_, sim[i])
          + rank_comp(sorted + 2 * N_, noi[i]);
  scores[i] = s;
}

// ---------------------------------------------------------------------------
// top-K (K=128): iterative argmax, ties -> lowest index (matches lax.top_k)
// ---------------------------------------------------------------------------
__global__ __launch_bounds__(1024) void topk_kernel(float* __restrict__ scores,
                                                    int* __restrict__ idxK) {
  __shared__ float ss[1024];
  __shared__ int   si[1024];
  int t = threadIdx.x;
  for (int r = 0; r < K_; ++r) {
    float bs = -1e30f; int bi = 0x7fffffff;
    for (int i = t; i < N_; i += 1024) {
      float v = scores[i];
      if (v > bs || (v == bs && i < bi)) { bs = v; bi = i; }
    }
    ss[t] = bs; si[t] = bi; __syncthreads();
    for (int o = 512; o > 0; o >>= 1) {
      if (t < o) {
        float v = ss[t + o]; int j = si[t + o];
        if (v > ss[t] || (v == ss[t] && j < si[t])) { ss[t] = v; si[t] = j; }
      }
      __syncthreads();
    }
    if (t == 0) { idxK[r] = si[0]; scores[si[0]] = -1e30f; }
    __syncthreads();
  }
}

// ---------------------------------------------------------------------------
// action embeddings: out[a] = W2 @ gelu(W1[:,a] + b1) + b2 (13 distinct actions)
// ---------------------------------------------------------------------------
__global__ __launch_bounds__(1024) void action_kernel(
    const float* __restrict__ mw1, const float* __restrict__ mb1,
    const float* __restrict__ mw2, const float* __restrict__ mb2,
    const float* __restrict__ tw1, const float* __restrict__ tb1,
    const float* __restrict__ tw2, const float* __restrict__ tb2,
    float* __restrict__ moveout, float* __restrict__ turnout) {
  __shared__ float g[H_];
  int a = blockIdx.x, t = threadIdx.x;
  if (a < 9) {
    g[t] = gelu_f(mw1[t * 9 + a] + mb1[t]);
    __syncthreads();
    float s = mb2[t];
    for (int h = 0; h < H_; ++h) s += mw2[(long)t * H_ + h] * g[h];
    moveout[a * H_ + t] = s;
  } else {
    int a2 = a - 9;
    g[t] = gelu_f(tw1[t * 4 + a2] + tb1[t]);
    __syncthreads();
    float s = tb2[t];
    for (int h = 0; h < H_; ++h) s += tw2[(long)t * H_ + h] * g[h];
    turnout[a2 * H_ + t] = s;
  }
}

// ---------------------------------------------------------------------------
// assemble token sequence: [start, v0, a0, v1, a1, ..., x] + pos_emb
// ---------------------------------------------------------------------------
__global__ void build_vecs_kernel(const float* __restrict__ start, const float* __restrict__ x,
                                  const float* __restrict__ mem, const int* __restrict__ idxK,
                                  const int* __restrict__ aim, const int* __restrict__ ait,
                                  const float* __restrict__ moveout, const float* __restrict__ turnout,
                                  const float* __restrict__ pos, float* __restrict__ vecs) {
  int s = blockIdx.x;
  int c = blockIdx.y * 256 + threadIdx.x;
  float v;
  if (s == 0) v = start[c];
  else if (s == S_ - 1) v = x[c];
  else if (s & 1) { int k = (s - 1) >> 1; v = mem[(long)idxK[k] * H_ + c]; }
  else {
    int k = (s >> 1) - 1; int m = idxK[k];
    v = 0.5f * (moveout[aim[m] * H_ + c] + turnout[ait[m] * H_ + c]);
  }
  vecs[(long)s * H_ + c] = v + pos[(long)s * H_ + c];
}

// ---------------------------------------------------------------------------
// layernorm over H=1024, one block per row
// ---------------------------------------------------------------------------
__global__ __launch_bounds__(256) void ln_kernel(const float* __restrict__ in,
                                                 const float* __restrict__ w,
                                                 const float* __restrict__ b,
                                                 float* __restrict__ out) {
  __shared__ float red[256];
  int s = blockIdx.x, t = threadIdx.x;
  const float* row = in + (long)s * H_;
  float sum = 0.f;
  for (int i = t; i < H_; i += 256) sum += row[i];
  red[t] = sum; __syncthreads();
  for (int o = 128; o > 0; o >>= 1) { if (t < o) red[t] += red[t + o]; __syncthreads(); }
  float mean = red[0] * (1.0f / H_); __syncthreads();
  float vs = 0.f;
  for (int i = t; i < H_; i += 256) { float d = row[i] - mean; vs += d * d; }
  red[t] = vs; __syncthreads();
  for (int o = 128; o > 0; o >>= 1) { if (t < o) red[t] += red[t + o]; __syncthreads(); }
  float inv = rsqrtf(red[0] * (1.0f / H_) + 1e-5f);
  for (int i = t; i < H_; i += 256)
    out[(long)s * H_ + i] = (row[i] - mean) * inv * w[i] + b[i];
}

// ---------------------------------------------------------------------------
// WMMA fp32 GEMM, TDM-staged + double-buffered LDS.
//   C = A (MxK) * op(B) + bias (+C), batched via blockIdx.y strides.
//   BT=true : B is (N x K) row-major -> C = A * B^T   (nn.Linear / Q@K^T)
//   BT=false: B is (K x N) row-major -> C = A * B     (attention @ V)
// Block: 256 threads = 8 waves; macro-tile 64(M) x 128(N); wave = 16x64 strip.
// Per K-chunk (KC_=32): wave0 issues 2 TDM descriptors for the *next* chunk,
// everyone computes the current chunk out of LDS, then s_wait_tensorcnt(0) +
// barrier. OOB rows/cols zero-fill via tensor_dim (no clamping needed).
// Requires K % 32 == 0 (all call sites satisfy this; attention K padded to 288).
// ---------------------------------------------------------------------------
template <bool BT, bool BIAS, bool ACT, bool ACCUM>
__global__ __launch_bounds__(256) void gemm_kernel(
    const float* __restrict__ A, int lda, long sA,
    const float* __restrict__ B, int ldb, long sB,
    const float* __restrict__ bias,
    float* __restrict__ C, int ldc, long sC,
    int M, int N, int K) {
  __shared__ __align__(16) float lds[2 * BUF_LDS];
  A += (long)blockIdx.y * sA;
  B += (long)blockIdx.y * sB;
  C += (long)blockIdx.y * sC;
  const int NB = (N + 127) >> 7;
  const int bm = blockIdx.x / NB, bn = blockIdx.x % NB;
  const int m0 = bm * 64, n0 = bn * 128;
  const int wv = threadIdx.x >> 5;
  const int msub = wv >> 1, nsub = wv & 1;
  const int lane = threadIdx.x & 31;
  const int half = lane >> 4, l16 = lane & 15;
  const int aml = msub * 16 + l16;                 // local A row for this lane
  const unsigned ldsbase = (unsigned)(size_t)(&lds[0]);
  const unsigned vm = (unsigned)((M - m0) < 64 ? (M - m0) : 64);
  const unsigned vn = (unsigned)((N - n0) < 128 ? (N - n0) : 128);
  const int nchunks = K / KC_;

  // prologue: stage chunk 0 into buffer 0
  if (wv == 0) {
    tdm_load_2d(ldsbase, A + (long)m0 * lda, KC_, vm, KC_, 64, (unsigned)lda, true);
    if (BT)
      tdm_load_2d(ldsbase + A_LDS * 4, B + (long)n0 * ldb, KC_, vn, KC_, 128,
                  (unsigned)ldb, true);
    else
      tdm_load_2d(ldsbase + A_LDS * 4, B + n0, vn, KC_, 128, KC_, (unsigned)ldb, false);
  }

  v8f z = {};
  v8f acc[4];
#pragma unroll
  for (int t = 0; t < 4; ++t) acc[t] = z;

  for (int c = 0; c < nchunks; ++c) {
    __builtin_amdgcn_s_wait_tensorcnt(0);  // wave0: chunk c landed; others: no-op
    __syncthreads();
    const int cur = c & 1;
    if (c + 1 < nchunks && wv == 0) {      // stage chunk c+1 while computing c
      const int k1 = (c + 1) * KC_;
      const unsigned nb = (unsigned)((c + 1) & 1) * BUF_LDS * 4u;
      tdm_load_2d(ldsbase + nb, A + (long)m0 * lda + k1, KC_, vm, KC_, 64,
                  (unsigned)lda, true);
      if (BT)
        tdm_load_2d(ldsbase + nb + A_LDS * 4, B + (long)n0 * ldb + k1, KC_, vn,
                    KC_, 128, (unsigned)ldb, true);
      else
        tdm_load_2d(ldsbase + nb + A_LDS * 4, B + (long)k1 * ldb + n0, vn, KC_,
                    128, KC_, (unsigned)ldb, false);
    }
    const float* LA = &lds[cur * BUF_LDS];
    const float* LB = LA + A_LDS;
#pragma unroll
    for (int kk = 0; kk < KC_; kk += 4) {
      const int ka = kk + 2 * half;
      v2f a = *(const v2f*)(LA + aml * APAD_ + ka);
#pragma unroll
      for (int t = 0; t < 4; ++t) {
        const int nl = nsub * 64 + t * 16 + l16;  // local B col
        v2f b;
        if (BT) {
          b = *(const v2f*)(LB + nl * APAD_ + ka);
        } else {
          b.x = LB[ka * 128 + nl];
          b.y = LB[(ka + 1) * 128 + nl];
        }
        acc[t] = __builtin_amdgcn_wmma_f32_16x16x4_f32(false, a, false, b,
                                                       (short)0, acc[t], false, false);
      }
    }
    __syncthreads();  // all reads of this buffer done before it is re-staged
  }

  // epilogue
#pragma unroll
  for (int t = 0; t < 4; ++t) {
    const int col = n0 + nsub * 64 + t * 16 + l16;
    if (col >= N) continue;
    float bv = BIAS ? bias[col] : 0.0f;
#pragma unroll
    for (int r = 0; r < 8; ++r) {
      const int row = m0 + msub * 16 + r + half * 8;
      if (row >= M) continue;
      float v = acc[t][r] + bv;
      if (ACT) v = gelu_f(v);
      float* cp = C + (long)row * ldc + col;
      if (ACCUM) v += *cp;
      *cp = v;
    }
  }
}

// ---------------------------------------------------------------------------
// causal softmax over att[h][q][0..q], scale 1/sqrt(64); zero-pads row to SK_
// ---------------------------------------------------------------------------
__global__ __launch_bounds__(256) void softmax_kernel(float* __restrict__ att) {
  __shared__ float red[256];
  int h = blockIdx.x / S_;
  int q = blockIdx.x % S_;
  float* row = att + ((long)h * S_ + q) * SK_;
  int t = threadIdx.x;
  float mx = -1e30f;
  for (int k = t; k <= q; k += 256) mx = fmaxf(mx, row[k] * 0.125f);
  red[t] = mx; __syncthreads();
  for (int o = 128; o > 0; o >>= 1) { if (t < o) red[t] = fmaxf(red[t], red[t + o]); __syncthreads(); }
  mx = red[0]; __syncthreads();
  float sum = 0.f;
  for (int k = t; k <= q; k += 256) sum += expf(row[k] * 0.125f - mx);
  red[t] = sum; __syncthreads();
  for (int o = 128; o > 0; o >>= 1) { if (t < o) red[t] += red[t + o]; __syncthreads(); }
  float inv = 1.0f / red[0]; __syncthreads();
  for (int k = t; k < SK_; k += 256)
    row[k] = (k <= q) ? expf(row[k] * 0.125f - mx) * inv : 0.0f;
}

// ---------------------------------------------------------------------------
// value head second layer: vp[i] = vh_b2 + dot(vh_mid[i], vh_w2)  (N=1 matvec)
// ---------------------------------------------------------------------------
__global__ __launch_bounds__(256) void vh2_kernel(const float* __restrict__ mid,
                                                  const float* __restrict__ w2,
                                                  const float* __restrict__ b2,
                                                  float* __restrict__ vp) {
  __shared__ float red[256];
  int i = blockIdx.x, t = threadIdx.x;
  float s = 0.f;
  for (int j = t; j < 4 * H_; j += 256) s += mid[(long)i * 4 * H_ + j] * w2[j];
  red[t] = s; __syncthreads();
  for (int o = 128; o > 0; o >>= 1) { if (t < o) red[t] += red[t + o]; __syncthreads(); }
  if (t == 0) vp[i] = red[0] + b2[0];
}

// ---------------------------------------------------------------------------
// pack outputs: vecs[1:], memory_next_vectors[idx], vp[-1], losses, sel_rewards
// ---------------------------------------------------------------------------
__global__ void finalize_kernel(const float* __restrict__ vecs,
                                const float* __restrict__ memnext,
                                const int* __restrict__ idxK,
                                const float* __restrict__ vp,
                                const float* __restrict__ rewards,
                                float* __restrict__ out, int total) {
  int i = blockIdx.x * 256 + threadIdx.x;
  if (i >= total) return;
  const int n1 = (S_ - 1) * H_;
  const int n2 = n1 + K_ * H_;
  if (i < n1) {
    out[i] = vecs[H_ + i];
  } else if (i < n2) {
    int j = i - n1; int k = j >> 10; int c = j & 1023;
    out[i] = memnext[(long)idxK[k] * H_ + c];
  } else {
    int j = i - n2;
    if (j == 0) out[i] = vp[K_];
    else if (j <= K_) { int m = j - 1; float d = vp[m] - rewards[idxK[m]]; out[i] = d * d; }
    else out[i] = rewards[idxK[j - 1 - K_]];
  }
}

// ===========================================================================
extern "C" void kernel_launch(void* const* d_in, const int* in_sizes, int n_in,
                              void* d_out, int out_size, void* d_ws, size_t ws_size,
                              hipStream_t stream) {
  (void)in_sizes; (void)n_in; (void)out_size; (void)ws_size;
  const float* x       = (const float*)d_in[0];
  const float* mem     = (const float*)d_in[1];
  const float* memnext = (const float*)d_in[2];
  const float* sur     = (const float*)d_in[3];
  const float* rew     = (const float*)d_in[4];
  const float* noise   = (const float*)d_in[5];
  const int*   aim     = (const int*)d_in[6];
  const int*   ait     = (const int*)d_in[7];
  const float* start   = (const float*)d_in[8];
  const float* pos     = (const float*)d_in[9];
  const float* ipw     = (const float*)d_in[10];
  const float* ipb     = (const float*)d_in[11];
  const float* ow      = (const float*)d_in[12];
  const float* ob      = (const float*)d_in[13];
  const float* l1w     = (const float*)d_in[14];
  const float* l1b     = (const float*)d_in[15];
  const float* l2w     = (const float*)d_in[16];
  const float* l2b     = (const float*)d_in[17];
  const float* f1w     = (const float*)d_in[18];
  const float* f1b     = (const float*)d_in[19];
  const float* f2w     = (const float*)d_in[20];
  const float* f2b     = (const float*)d_in[21];
  const float* amw1    = (const float*)d_in[22];
  const float* amb1    = (const float*)d_in[23];
  const float* amw2    = (const float*)d_in[24];
  const float* amb2    = (const float*)d_in[25];
  const float* atw1    = (const float*)d_in[26];
  const float* atb1    = (const float*)d_in[27];
  const float* atw2    = (const float*)d_in[28];
  const float* atb2    = (const float*)d_in[29];
  const float* vhw1    = (const float*)d_in[30];
  const float* vhb1    = (const float*)d_in[31];
  const float* vhw2    = (const float*)d_in[32];
  const float* vhb2    = (const float*)d_in[33];
  float* out = (float*)d_out;

  // ---- workspace carve-up (floats) ----
  float* W        = (float*)d_ws;
  float* w_sorted = W;                             // 3*N
  float* w_scores = w_sorted + 3L * N_;            // N
  float* w_sim    = w_scores + N_;                 // N
  float* w_xnorm  = w_sim + N_;                    // 32
  int*   w_idx    = (int*)(w_xnorm + 32);          // 128 ints
  float* w_move   = w_xnorm + 32 + 128;            // 9*H
  float* w_turn   = w_move + 9 * H_;               // 4*H
  float* w_vecs   = w_turn + 4 * H_;               // S*H
  float* w_ln     = w_vecs + (long)S_ * H_;        // S*H
  float* w_qkv    = w_ln + (long)S_ * H_;          // S*3H
  float* w_att    = w_qkv + (long)S_ * 3 * H_;     // NH*S*SK
  float* w_attout = w_att + (long)NH_ * S_ * SK_;  // S*H
  float* w_ffn    = w_attout + (long)S_ * H_;      // S*4H
  float* w_vhmid  = w_ffn + (long)S_ * 4 * H_;     // 129*4H
  float* w_vp     = w_vhmid + 129L * 4 * H_;       // 129

  // ---- scoring ----
  xnorm_kernel<<<1, 256, 0, stream>>>(x, w_xnorm);
  sim_kernel<<<N_ / 8, 256, 0, stream>>>(mem, x, w_xnorm, w_sim);
  copy3_kernel<<<N_ / 256, 256, 0, stream>>>(sur, w_sim, noise, w_sorted);
  {
    dim3 sg(N_ / 256, 3);
    for (int k = 2; k <= N_; k <<= 1)
      for (int j = k >> 1; j > 0; j >>= 1)
        bitonic_pass<<<sg, 256, 0, stream>>>(w_sorted, k, j);
  }
  rank_kernel<<<N_ / 256, 256, 0, stream>>>(sur, w_sim, noise, w_sorted, w_scores);
  topk_kernel<<<1, 1024, 0, stream>>>(w_scores, w_idx);

  // ---- action embeddings + sequence assembly ----
  action_kernel<<<13, 1024, 0, stream>>>(amw1, amb1, amw2, amb2,
                                         atw1, atb1, atw2, atb2, w_move, w_turn);
  build_vecs_kernel<<<dim3(S_, H_ / 256), 256, 0, stream>>>(
      start, x, mem, w_idx, aim, ait, w_move, w_turn, pos, w_vecs);

  auto ggrid = [](int M, int N, int batch) {
    int MB = (M + 63) / 64, NB = (N + 127) / 128;
    return dim3((unsigned)(MB * NB), (unsigned)batch);
  };

  // ---- transformer (L=4) ----
  for (int l = 0; l < L_; ++l) {
    const float* ipw_l = ipw + (long)l * 3 * H_ * H_;
    const float* ipb_l = ipb + (long)l * 3 * H_;
    const float* ow_l  = ow  + (long)l * H_ * H_;
    const float* ob_l  = ob  + (long)l * H_;
    const float* f1w_l = f1w + (long)l * 4 * H_ * H_;
    const float* f1b_l = f1b + (long)l * 4 * H_;
    const float* f2w_l = f2w + (long)l * H_ * 4 * H_;
    const float* f2b_l = f2b + (long)l * H_;

    // LN1 -> QKV
    ln_kernel<<<S_, 256, 0, stream>>>(w_vecs, l1w + l * H_, l1b + l * H_, w_ln);
    gemm_kernel<true, true, false, false><<<ggrid(S_, 3 * H_, 1), 256, 0, stream>>>(
        w_ln, H_, 0, ipw_l, H_, 0, ipb_l, w_qkv, 3 * H_, 0, S_, 3 * H_, H_);
    // att = Q @ K^T per head  (K=DH_=64)
    gemm_kernel<true, false, false, false><<<ggrid(S_, S_, NH_), 256, 0, stream>>>(
        w_qkv, 3 * H_, DH_, w_qkv + H_, 3 * H_, DH_, (const float*)nullptr,
        w_att, SK_, (long)S_ * SK_, S_, S_, DH_);
    softmax_kernel<<<NH_ * S_, 256, 0, stream>>>(w_att);
    // O = att @ V per head  (K padded to SK_ with zeros)
    gemm_kernel<false, false, false, false><<<ggrid(S_, DH_, NH_), 256, 0, stream>>>(
        w_att, SK_, (long)S_ * SK_, w_qkv + 2 * H_, 3 * H_, DH_, (const float*)nullptr,
        w_attout, H_, DH_, S_, DH_, SK_);
    // out proj + residual
    gemm_kernel<true, true, false, true><<<ggrid(S_, H_, 1), 256, 0, stream>>>(
        w_attout, H_, 0, ow_l, H_, 0, ob_l, w_vecs, H_, 0, S_, H_, H_);
    // LN2 -> FFN
    ln_kernel<<<S_, 256, 0, stream>>>(w_vecs, l2w + l * H_, l2b + l * H_, w_ln);
    gemm_kernel<true, true, true, false><<<ggrid(S_, 4 * H_, 1), 256, 0, stream>>>(
        w_ln, H_, 0, f1w_l, H_, 0, f1b_l, w_ffn, 4 * H_, 0, S_, 4 * H_, H_);
    gemm_kernel<true, true, false, true><<<ggrid(S_, H_, 1), 256, 0, stream>>>(
        w_ffn, 4 * H_, 0, f2w_l, 4 * H_, 0, f2b_l, w_vecs, H_, 0, S_, H_, 4 * H_);
  }

  // ---- value head: rows vecs[1::2] expressed via lda = 2H ----
  gemm_kernel<true, true, true, false><<<ggrid(K_ + 1, 4 * H_, 1), 256, 0, stream>>>(
      w_vecs + H_, 2 * H_, 0, vhw1, H_, 0, vhb1, w_vhmid, 4 * H_, 0, K_ + 1, 4 * H_, H_);
  vh2_kernel<<<K_ + 1, 256, 0, stream>>>(w_vhmid, vhw2, vhb2, w_vp);

  // ---- pack outputs ----
  const int total = (S_ - 1) * H_ + K_ * H_ + 1 + K_ + K_;
  finalize_kernel<<<(total + 255) / 256, 256, 0, stream>>>(
      w_vecs, memnext, w_idx, w_vp, rew, out, total);
}